// DWT_1090921693502
// MI455X (gfx1250) — compile-verified
//
#include <hip/hip_runtime.h>
#include <hip/hip_bf16.h>

// ---------------------------------------------------------------------------
// Haar DWT2 on (8,64,512,512) fp32 -> 4x (8,64,256,256) fp32 (ll,lh,hl,hh).
// Pure streaming: 1 GiB total traffic, AI ~0.25 flop/byte -> HBM-bound
// (~46 us floor @ 23.3 TB/s). Strategy: b128 non-temporal loads/stores,
// full coalescing, shift/mask indexing, max wave parallelism.
// ---------------------------------------------------------------------------

typedef float v4f __attribute__((ext_vector_type(4)));
typedef int   v4i __attribute__((ext_vector_type(4)));

#define W_SHIFT   9    // W  = 512
#define IMG_SHIFT 18   // H*W = 2^18
#define OW_SHIFT  8    // OW = 256
#define OIMG_SHIFT 16  // OH*OW = 2^16

__global__ __launch_bounds__(256) void haar_dwt2_kernel(
    const float* __restrict__ x,
    float* __restrict__ oll, float* __restrict__ olh,
    float* __restrict__ ohl, float* __restrict__ ohh,
    unsigned nthreads)
{
    unsigned t = blockIdx.x * 256u + threadIdx.x;
    if (t >= nthreads) return;

    // Each thread: 4 output pixels (one row, 4 consecutive cols)
    //   gj  in [0, 64)  : group of 4 output cols  (8 input cols)
    //   oi  in [0, 256) : output row
    //   img in [0, 512) : image index (B*C)
    unsigned gj  = t & 63u;
    unsigned oi  = (t >> 6) & 255u;
    unsigned img = t >> 14;

    size_t ibase = ((size_t)img << IMG_SHIFT)
                 + ((size_t)(oi << 1) << W_SHIFT)
                 + ((size_t)gj << 3);

    const v4f* p0 = (const v4f*)(x + ibase);          // input row 2*oi
    const v4f* p1 = (const v4f*)(x + ibase + 512);    // input row 2*oi+1

    v4f r0a = __builtin_nontemporal_load(p0);
    v4f r0b = __builtin_nontemporal_load(p0 + 1);
    v4f r1a = __builtin_nontemporal_load(p1);
    v4f r1b = __builtin_nontemporal_load(p1 + 1);

    v4f vll, vlh, vhl, vhh;

#define HAAR_QUAD(a00, a01, a10, a11, F)                         \
    {                                                            \
        float s0 = (a00) + (a01);                                \
        float d0 = (a00) - (a01);                                \
        float s1 = (a10) + (a11);                                \
        float d1 = (a10) - (a11);                                \
        vll.F = 0.5f * (s0 + s1);                                \
        vlh.F = 0.5f * (s0 - s1);                                \
        vhl.F = 0.5f * (d0 + d1);                                \
        vhh.F = 0.5f * (d0 - d1);                                \
    }

    HAAR_QUAD(r0a.x, r0a.y, r1a.x, r1a.y, x);
    HAAR_QUAD(r0a.z, r0a.w, r1a.z, r1a.w, y);
    HAAR_QUAD(r0b.x, r0b.y, r1b.x, r1b.y, z);
    HAAR_QUAD(r0b.z, r0b.w, r1b.z, r1b.w, w);
#undef HAAR_QUAD

    size_t obase = ((size_t)img << OIMG_SHIFT)
                 + ((size_t)oi << OW_SHIFT)
                 + ((size_t)gj << 2);

    __builtin_nontemporal_store(vll, (v4f*)(oll + obase));
    __builtin_nontemporal_store(vlh, (v4f*)(olh + obase));
    __builtin_nontemporal_store(vhl, (v4f*)(ohl + obase));
    __builtin_nontemporal_store(vhh, (v4f*)(ohh + obase));
}

// ---------------------------------------------------------------------------
// Compile-probe (never launched): gfx1250 async-to-LDS path. Round-1 stderr
// confirmed the builtin exists and expects (int4 AS1*, int4 AS3*, Imm, Imm)
// -- clang prints address_space(1) as "__device__" in HIP mode.
// ---------------------------------------------------------------------------
#if defined(__gfx1250__) && defined(__has_builtin)
#if __has_builtin(__builtin_amdgcn_global_load_async_to_lds_b128) && \
    __has_builtin(__builtin_amdgcn_s_wait_asynccnt)
#define HAVE_CDNA5_ASYNC_LDS 1
#endif
#endif

#ifdef HAVE_CDNA5_ASYNC_LDS
typedef __attribute__((address_space(1))) v4i gv4i;
typedef __attribute__((address_space(3))) v4i lv4i;

__global__ __launch_bounds__(32) void cdna5_async_probe_kernel(
    const float* __restrict__ g, float* __restrict__ out)
{
    __shared__ __align__(16) float lds[128];
    gv4i* gsrc = (gv4i*)(g + threadIdx.x * 4u);
    lv4i* ldst = (lv4i*)&lds[threadIdx.x * 4u];
    __builtin_amdgcn_global_load_async_to_lds_b128(gsrc, ldst, 0, 0);
    __builtin_amdgcn_s_wait_asynccnt(0);
    __syncthreads();
    out[threadIdx.x] = lds[threadIdx.x];
}
#endif

// ---------------------------------------------------------------------------

extern "C" void kernel_launch(void* const* d_in, const int* in_sizes, int n_in,
                              void* d_out, int out_size, void* d_ws, size_t ws_size,
                              hipStream_t stream) {
    (void)n_in; (void)out_size; (void)d_ws; (void)ws_size;

    const float* x = (const float*)d_in[0];
    float* out = (float*)d_out;

    size_t total = (size_t)in_sizes[0];   // 8*64*512*512 = 134,217,728
    size_t nout  = total >> 2;            // per-plane output elements

    float* oll = out;
    float* olh = out + nout;
    float* ohl = out + 2 * nout;
    float* ohh = out + 3 * nout;

    unsigned nthreads = (unsigned)(nout >> 2);   // 4 output px per thread
    unsigned blocks   = (nthreads + 255u) / 256u;

    haar_dwt2_kernel<<<blocks, 256, 0, stream>>>(x, oll, olh, ohl, ohh, nthreads);
}